// AEVSV_43095701848237
// MI455X (gfx1250) — compile-verified
//
#include <hip/hip_runtime.h>
#include <math.h>

#define BB   16
#define NN   256
#define NSH  16

typedef __attribute__((ext_vector_type(2))) float v2f;
typedef __attribute__((ext_vector_type(4))) float v4f;
typedef __attribute__((ext_vector_type(8))) float v8f;

// -----------------------------------------------------------------------------
// Kernel 1: d_ij via V_WMMA_F32_16X16X4_F32 (Gram-matrix trick).
// One wave (32 lanes) per 16x16 (i,j) tile. K=4 (x,y,z,0).
//   A layout (16x4 f32): lanes 0-15 hold row M=lane: K0 in v0, K1 in v1;
//                        lanes 16-31 hold row M=lane-16: K2 in v0, K3 in v1.
//   B layout mirrors A with columns (N) in place of rows.
//   C/D layout: VGPR r: lanes 0-15 -> (M=r, N=lane); lanes 16-31 -> (M=r+8).
// -----------------------------------------------------------------------------
__global__ __launch_bounds__(32) void aev_dmat_wmma(
    const float* __restrict__ coord, const float* __restrict__ rc_p,
    float* __restrict__ d_mat)
{
    const float rc = rc_p[0];
    const int b  = blockIdx.z;
    const int ti = blockIdx.y;
    const int tj = blockIdx.x;
    const int l  = threadIdx.x;
    const int lo = l & 15;
    const int hi = l >> 4;

    const float* cb = coord + (size_t)b * NN * 3;
    const int row = ti * 16 + lo;
    const int col = tj * 16 + lo;
    const float rix = cb[row * 3 + 0], riy = cb[row * 3 + 1], riz = cb[row * 3 + 2];
    const float cjx = cb[col * 3 + 0], cjy = cb[col * 3 + 1], cjz = cb[col * 3 + 2];

    v2f a, bm;
    if (hi) { a.x = riz; a.y = 0.0f; bm.x = cjz; bm.y = 0.0f; }
    else    { a.x = rix; a.y = riy;  bm.x = cjx; bm.y = cjy;  }

    v8f c = {};
    // D = A * B + 0  -> 16x16 tile of dot(ri, rj)
    v8f g = __builtin_amdgcn_wmma_f32_16x16x4_f32(
        /*neg_a=*/false, a, /*neg_b=*/false, bm,
        /*c_mod=*/(short)0, c, /*reuse_a=*/false, /*reuse_b=*/false);

    __shared__ float rn[16];  // ||r_i||^2 for the 16 rows of this tile
    __shared__ float cn[16];  // ||r_j||^2 for the 16 cols of this tile
    if (hi == 0) rn[lo] = rix * rix + riy * riy + riz * riz;
    else         cn[lo] = cjx * cjx + cjy * cjy + cjz * cjz;
    __syncthreads();

    float* dbase = d_mat + (size_t)b * NN * NN;
    #pragma unroll
    for (int r = 0; r < 8; ++r) {
        const int M = r + 8 * hi;
        const int i = ti * 16 + M;
        const int j = tj * 16 + lo;
        const float sq = rn[M] + cn[lo] - 2.0f * g[r];
        // diagonal: reference fills with rc_s; clamp guards tiny negative sq
        const float d = (i == j) ? rc : sqrtf(fmaxf(sq, 0.0f));
        dbase[(size_t)i * NN + j] = d;   // cacheable: re-read by kernel 2 from L2
    }
}

// -----------------------------------------------------------------------------
// Kernel 2: bandwidth-dominated gs/gv production (268 MB of streaming stores).
// One 256-thread block per (b, i) row. Point coords + shifts staged in LDS.
// Thread t, iteration it: slot = it*256+t, j = slot>>2, s4 = slot&3 -> computes
// shifts [4*s4, 4*s4+4) and emits 4x 128-bit NON-TEMPORAL stores (1 gs, 3 gv),
// fully coalesced (gs vector-index == slot).
// -----------------------------------------------------------------------------
__global__ __launch_bounds__(256) void aev_gsgv(
    const float* __restrict__ coord, const float* __restrict__ shifts,
    const float* __restrict__ eta_p, const float* __restrict__ rc_p,
    const float* __restrict__ d_mat,
    float* __restrict__ gs_out, float* __restrict__ gv_out)
{
    __shared__ float scj[NN * 3];
    __shared__ float ssh[NSH];

    const int b = blockIdx.y;
    const int i = blockIdx.x;
    const int t = threadIdx.x;

    const float* cb = coord + (size_t)b * NN * 3;
    #pragma unroll
    for (int k = t; k < NN * 3; k += 256) scj[k] = cb[k];
    if (t < NSH) ssh[t] = shifts[t];
    __syncthreads();

    const float eta = eta_p[0];
    const float rc  = rc_p[0];
    const float cix = scj[i * 3 + 0];
    const float ciy = scj[i * 3 + 1];
    const float ciz = scj[i * 3 + 2];
    const float pi_over_rc = 3.14159265358979323846f / rc;

    const float* drow = d_mat + ((size_t)b * NN + i) * NN;
    v4f* gsrow = (v4f*)(gs_out + ((size_t)b * NN + i) * NN * NSH);
    v4f* gvrow = (v4f*)(gv_out + ((size_t)b * NN + i) * NN * 3 * NSH);

    #pragma unroll
    for (int it = 0; it < 4; ++it) {
        const int slot = it * 256 + t;
        const int j  = slot >> 2;   // pair column
        const int s4 = slot & 3;    // which group of 4 shifts

        const float d  = drow[j];   // WMMA-produced distance (L2 hit)
        const float fc = (d < rc) ? (0.5f * __cosf(pi_over_rc * d) + 0.5f) : 0.0f;
        const float inv = 1.0f / d;
        const float ux = (scj[j * 3 + 0] - cix) * inv;
        const float uy = (scj[j * 3 + 1] - ciy) * inv;
        const float uz = (scj[j * 3 + 2] - ciz) * inv;

        v4f g;
        #pragma unroll
        for (int q = 0; q < 4; ++q) {
            const float dt = d - ssh[s4 * 4 + q];
            g[q] = __expf(-eta * dt * dt) * fc;
        }
        const v4f gx = g * ux;
        const v4f gy = g * uy;
        const v4f gz = g * uz;

        // streaming outputs: never re-read -> bypass cache retention (NT)
        __builtin_nontemporal_store(g,  &gsrow[j * 4 + s4]);
        __builtin_nontemporal_store(gx, &gvrow[j * 12 + 0 * 4 + s4]);
        __builtin_nontemporal_store(gy, &gvrow[j * 12 + 1 * 4 + s4]);
        __builtin_nontemporal_store(gz, &gvrow[j * 12 + 2 * 4 + s4]);
    }
}

extern "C" void kernel_launch(void* const* d_in, const int* in_sizes, int n_in,
                              void* d_out, int out_size, void* d_ws, size_t ws_size,
                              hipStream_t stream) {
    (void)in_sizes; (void)n_in; (void)out_size; (void)d_ws; (void)ws_size;

    const float* coord   = (const float*)d_in[0];  // [B,N,3]
    const float* shifts  = (const float*)d_in[1];  // [16]
    const float* eta_p   = (const float*)d_in[2];  // scalar
    const float* rc_p    = (const float*)d_in[3];  // scalar

    float* out  = (float*)d_out;
    float* dmat = out;                                           // [B,N,N]
    float* gs   = out + (size_t)BB * NN * NN;                    // [B,N,N,16]
    float* gv   = gs  + (size_t)BB * NN * NN * NSH;              // [B,N,N,3,16]

    // Kernel 1: 16x16x16 = 4096 waves, one 16x16 tile each (WMMA path)
    dim3 g1(NN / 16, NN / 16, BB);
    aev_dmat_wmma<<<g1, 32, 0, stream>>>(coord, rc_p, dmat);

    // Kernel 2: one block per (b,i) row; 268 MB of NT streaming stores
    dim3 g2(NN, BB);
    aev_gsgv<<<g2, 256, 0, stream>>>(coord, shifts, eta_p, rc_p, dmat, gs, gv);
}